// RRL_20040317403436
// MI455X (gfx1250) — compile-verified
//
#include <hip/hip_runtime.h>

// LBP min-rotation selector over non-overlapping 3x3 windows.
// Memory-bound streaming kernel: 604 MB total traffic, ~26us floor @ 23.3 TB/s.
// WMMA does not apply (no contraction; data-dependent permutation select).
// Strategy: 4 windows per thread -> all memory ops are aligned b128, coalesced,
// with non-temporal hints (data is touched exactly once; don't churn L2).

typedef float v4f __attribute__((ext_vector_type(4)));

#define IMG_W 384
#define OH 128
#define GROUPS_PER_ROW 32  // 128 windows/row, 4 windows per thread-group

__global__ __launch_bounds__(256) void lbp_minrot_kernel(
    const float* __restrict__ x, float* __restrict__ out, int n_groups)
{
    int gid = blockIdx.x * 256 + threadIdx.x;
    if (gid >= n_groups) return;

    int g   = gid & (GROUPS_PER_ROW - 1);   // group of 4 windows within row
    int t   = gid >> 5;
    int oh  = t & (OH - 1);                 // window-row
    int img = t >> 7;                       // collapsed B*C image index

    size_t base = (size_t)img * (IMG_W * IMG_W) + (size_t)(3 * oh) * IMG_W + (size_t)g * 12;

    const v4f* p0 = (const v4f*)(x + base);
    const v4f* p1 = (const v4f*)(x + base + IMG_W);
    const v4f* p2 = (const v4f*)(x + base + 2 * IMG_W);

    float r0[12], r1[12], r2[12];
#pragma unroll
    for (int q = 0; q < 3; ++q) {
        v4f a = __builtin_nontemporal_load(p0 + q);
        v4f b = __builtin_nontemporal_load(p1 + q);
        v4f c = __builtin_nontemporal_load(p2 + q);
#pragma unroll
        for (int k = 0; k < 4; ++k) {
            r0[4 * q + k] = a[k];
            r1[4 * q + k] = b[k];
            r2[4 * q + k] = c[k];
        }
    }

    float o0[12], o1[12], o2[12];
#pragma unroll
    for (int j = 0; j < 4; ++j) {
        float w00 = r0[3 * j], w01 = r0[3 * j + 1], w02 = r0[3 * j + 2];
        float w10 = r1[3 * j], w11 = r1[3 * j + 1], w12 = r1[3 * j + 2];
        float w20 = r2[3 * j], w21 = r2[3 * j + 1], w22 = r2[3 * j + 2];

        // All four rotation keys packed into one u32 (one byte each).
        // Byte b of constant = raster LBP weight this neighbor gets under rotation b.
        unsigned kv = 0u;
        kv += (w00 >= w11) ? 0x04802001u : 0u;
        kv += (w01 >= w11) ? 0x10400802u : 0u;
        kv += (w02 >= w11) ? 0x80200104u : 0u;
        kv += (w10 >= w11) ? 0x02104008u : 0u;
        kv += (w12 >= w11) ? 0x40080210u : 0u;
        kv += (w20 >= w11) ? 0x01048020u : 0u;
        kv += (w21 >= w11) ? 0x08021040u : 0u;
        kv += (w22 >= w11) ? 0x20010480u : 0u;

        unsigned k0 = kv & 0xffu;
        unsigned k1 = (kv >> 8) & 0xffu;
        unsigned k2 = (kv >> 16) & 0xffu;
        unsigned k3 = kv >> 24;

        // argmin with first-occurrence tie-break (strict <)
        int idx = 0;
        unsigned best = k0;
        if (k1 < best) { best = k1; idx = 1; }
        if (k2 < best) { best = k2; idx = 2; }
        if (k3 < best) { best = k3; idx = 3; }
        bool i1 = (idx == 1), i2 = (idx == 2), i3 = (idx == 3);

        // rot1[i][j] = w[j][2-i] (np.rot90 CCW), rot2 = 180, rot3 = CW
        o0[3 * j]     = i1 ? w02 : (i2 ? w22 : (i3 ? w20 : w00));
        o0[3 * j + 1] = i1 ? w12 : (i2 ? w21 : (i3 ? w10 : w01));
        o0[3 * j + 2] = i1 ? w22 : (i2 ? w20 : (i3 ? w00 : w02));
        o1[3 * j]     = i1 ? w01 : (i2 ? w12 : (i3 ? w21 : w10));
        o1[3 * j + 1] = w11;  // center is rotation-invariant
        o1[3 * j + 2] = i1 ? w21 : (i2 ? w10 : (i3 ? w01 : w12));
        o2[3 * j]     = i1 ? w00 : (i2 ? w02 : (i3 ? w22 : w20));
        o2[3 * j + 1] = i1 ? w10 : (i2 ? w01 : (i3 ? w12 : w21));
        o2[3 * j + 2] = i1 ? w20 : (i2 ? w00 : (i3 ? w02 : w22));
    }

    v4f* q0 = (v4f*)(out + base);
    v4f* q1 = (v4f*)(out + base + IMG_W);
    v4f* q2 = (v4f*)(out + base + 2 * IMG_W);
#pragma unroll
    for (int q = 0; q < 3; ++q) {
        v4f a, b, c;
#pragma unroll
        for (int k = 0; k < 4; ++k) {
            a[k] = o0[4 * q + k];
            b[k] = o1[4 * q + k];
            c[k] = o2[4 * q + k];
        }
        __builtin_nontemporal_store(a, q0 + q);
        __builtin_nontemporal_store(b, q1 + q);
        __builtin_nontemporal_store(c, q2 + q);
    }
}

extern "C" void kernel_launch(void* const* d_in, const int* in_sizes, int n_in,
                              void* d_out, int out_size, void* d_ws, size_t ws_size,
                              hipStream_t stream) {
    const float* x = (const float*)d_in[0];
    float* out = (float*)d_out;

    long long n = (long long)in_sizes[0];
    int n_images = (int)(n / ((long long)IMG_W * IMG_W));   // B*C = 512
    int n_groups = n_images * OH * GROUPS_PER_ROW;          // 2,097,152

    int block = 256;
    int grid = (n_groups + block - 1) / block;
    hipLaunchKernelGGL(lbp_minrot_kernel, dim3(grid), dim3(block), 0, stream,
                       x, out, n_groups);
}